// tNN4MNIST_19963007992237
// MI455X (gfx1250) — compile-verified
//
#include <hip/hip_runtime.h>

typedef __attribute__((ext_vector_type(2))) float v2f;
typedef __attribute__((ext_vector_type(4))) float v4f;
typedef __attribute__((ext_vector_type(8))) float v8f;

#define TN 16
#define LDC 448          // 28 * TN
#define ACT_ELEMS 12544  // 28 * LDC
#define NTHREADS 256
#define NWAVES 8

// ---------------- init kernels (tiny, run every launch) ----------------

// ws[0..895]   : C   (32 rows padded x 28), C[k][l]
// ws[896..1791]: C^T (32 rows padded x 28), CT[l][k] = C[k][l]
__global__ void init_dct(float* __restrict__ ws) {
  int idx = blockIdx.x * blockDim.x + threadIdx.x;
  if (idx >= 896) return;
  int r = idx / 28, c = idx % 28;
  const float PI = 3.14159265358979323846f;
  float cval = 0.f, tval = 0.f;
  if (r < 28) {
    float s0 = (r == 0) ? sqrtf(1.f / 28.f) : sqrtf(2.f / 28.f);
    cval = s0 * cosf(PI * (float)((2 * c + 1) * r) / 56.f);
    float s1 = (c == 0) ? sqrtf(1.f / 28.f) : sqrtf(2.f / 28.f);
    tval = s1 * cosf(PI * (float)((2 * r + 1) * c) / 56.f);
  }
  ws[idx] = cval;
  ws[896 + idx] = tval;
}

// dst[k][a][b] (28 x 32pad x 28) = sum_l C[k][l] * W[l][a][b], W is (28, arows, 28)
__global__ void init_what(const float* __restrict__ C, const float* __restrict__ W,
                          float* __restrict__ dst, int arows) {
  int idx = blockIdx.x * blockDim.x + threadIdx.x;
  if (idx >= 28 * 32 * 28) return;
  int k = idx / (32 * 28);
  int a = (idx / 28) % 32;
  int b = idx % 28;
  float acc = 0.f;
  if (a < arows) {
    for (int l = 0; l < 28; ++l)
      acc += C[k * 28 + l] * W[(l * arows + a) * 28 + b];
  }
  dst[idx] = acc;
}

// ---------------- WMMA helpers ----------------
// A frag: a[v] = A[m][kk*4 + 2*half + v]; B frag: b[v] = B[kk*4 + 2*half + v][n]
// D frag: element (v + 8*half (+row tile), col0 + lane%16) = acc[v]         (§7.12.2)

// Two M-tiles (rows 0-15 and 16-31) sharing the same B fragments.
__device__ __forceinline__ void mm2_k28(const float* __restrict__ Am,
                                        const float* Bm, int ldb, int ncol0,
                                        v8f& acc0, v8f& acc1) {
  const int lane = threadIdx.x & 31;
  const int l15 = lane & 15;
  const int half = lane >> 4;
  const float* ap0 = Am + (size_t)l15 * 28 + 2 * half;
  const float* ap1 = ap0 + 16 * 28;
  const float* bp = Bm + (size_t)(2 * half) * ldb + ncol0 + l15;
#pragma unroll
  for (int kk = 0; kk < 7; ++kk) {
    v2f b;
    b.x = bp[0];
    b.y = bp[ldb];
    v2f a0, a1;
    a0.x = ap0[0];
    a0.y = ap0[1];
    a1.x = ap1[0];
    a1.y = ap1[1];
    acc0 = __builtin_amdgcn_wmma_f32_16x16x4_f32(false, a0, false, b, (short)0, acc0,
                                                 false, false);
    acc1 = __builtin_amdgcn_wmma_f32_16x16x4_f32(false, a1, false, b, (short)0, acc1,
                                                 false, false);
    ap0 += 4;
    ap1 += 4;
    bp += 4 * ldb;
  }
}

// Single M-tile (rows 0-15).
__device__ __forceinline__ void mm1_k28(const float* __restrict__ Am,
                                        const float* Bm, int ldb, int ncol0, v8f& acc0) {
  const int lane = threadIdx.x & 31;
  const int l15 = lane & 15;
  const int half = lane >> 4;
  const float* ap0 = Am + (size_t)l15 * 28 + 2 * half;
  const float* bp = Bm + (size_t)(2 * half) * ldb + ncol0 + l15;
#pragma unroll
  for (int kk = 0; kk < 7; ++kk) {
    v2f b;
    b.x = bp[0];
    b.y = bp[ldb];
    v2f a0;
    a0.x = ap0[0];
    a0.y = ap0[1];
    acc0 = __builtin_amdgcn_wmma_f32_16x16x4_f32(false, a0, false, b, (short)0, acc0,
                                                 false, false);
    ap0 += 4;
    bp += 4 * ldb;
  }
}

// rows 0..15: always in bounds -> unconditional packed LDS stores
__device__ __forceinline__ void st16(float* dst, int ldd, int col0, v8f acc) {
  const int lane = threadIdx.x & 31;
  float* p = dst + (size_t)((lane >> 4) << 3) * ldd + col0 + (lane & 15);
#pragma unroll
  for (int v = 0; v < 8; ++v) p[v * ldd] = acc[v];
}

// rows 16..31, valid through 27: rows 28-31 (half==1, v>=4) redirected to dump word
__device__ __forceinline__ void sthi28(float* dst, int ldd, int col0, v8f acc,
                                       float* dump) {
  const int lane = threadIdx.x & 31;
  const int half = lane >> 4;
  float* p = dst + (size_t)(16 + (half << 3)) * ldd + col0 + (lane & 15);
#pragma unroll
  for (int v = 0; v < 4; ++v) p[v * ldd] = acc[v];
#pragma unroll
  for (int v = 4; v < 8; ++v) {
    float* q = half ? dump : (p + v * ldd);
    *q = acc[v];
  }
}

// ---------------- fused network kernel ----------------

__global__ void __launch_bounds__(NTHREADS) tnn_fused(
    const float* __restrict__ x, const float* __restrict__ ws,
    const float* __restrict__ B1, const float* __restrict__ B2,
    const float* __restrict__ B3, const float* __restrict__ B4,
    float* __restrict__ out, int N) {
  extern __shared__ float lds[];
  float* cur = lds;              // ACT_ELEMS floats
  float* oth = lds + ACT_ELEMS;  // ACT_ELEMS floats
  float* dump = lds + 2 * ACT_ELEMS;  // discard slot for padded rows

  const float* Cm  = ws;
  const float* CTm = ws + 896;

  const int n0 = blockIdx.x * TN;
  const int tid = threadIdx.x;
  const int wave = tid >> 5;

  // S0: load x tile (28 k x 28 b x 16 n) -> cur[k*448 + b*16 + n], b128 fast path
  if (((N & 3) == 0) && (n0 + TN <= N)) {
    for (int i4 = tid; i4 < (ACT_ELEMS / 4); i4 += NTHREADS) {
      int row = i4 >> 2;  // k*28 + b
      int sub = i4 & 3;
      v4f v = *reinterpret_cast<const v4f*>(x + (size_t)row * N + n0 + (sub << 2));
      *reinterpret_cast<v4f*>(cur + ((size_t)i4 << 2)) = v;
    }
  } else {
    for (int idx = tid; idx < ACT_ELEMS; idx += NTHREADS) {
      int nl = idx & 15;
      float v = 0.f;
      if (n0 + nl < N) v = x[(size_t)(idx >> 4) * N + n0 + nl];
      cur[idx] = v;
    }
  }
  __syncthreads();

  for (int layer = 0; layer < 4; ++layer) {
    const bool last = (layer == 3);
    const float* Wl = (layer == 0)   ? ws + 1792
                      : (layer == 1) ? ws + 26880
                      : (layer == 2) ? ws + 51968
                                     : ws + 77056;
    const float* Bl = (layer == 0) ? B1 : (layer == 1) ? B2 : (layer == 2) ? B3 : B4;
    const int ldY = last ? 256 : LDC;  // layer-4 Y slices padded to 16 rows

    // ---- S1: oth = C @ cur  (forward DCT; 32x28 @ 28x448, both M-tiles per nt) ----
    for (int nt = wave; nt < 28; nt += NWAVES) {
      v8f a0 = {}, a1 = {};
      mm2_k28(Cm, cur, LDC, nt * 16, a0, a1);
      st16(oth, LDC, nt * 16, a0);
      sthi28(oth, LDC, nt * 16, a1, dump);
    }
    __syncthreads();

    // ---- S2: cur[k][a][n] = What[k] @ oth[k]  (per-slice matmul) ----
    if (!last) {
      for (int k = wave; k < 28; k += NWAVES) {
        v8f a0 = {}, a1 = {};
        mm2_k28(Wl + (size_t)k * 896, oth + (size_t)k * LDC, TN, 0, a0, a1);
        st16(cur + (size_t)k * LDC, TN, 0, a0);
        sthi28(cur + (size_t)k * LDC, TN, 0, a1, dump);
      }
    } else {
      for (int k = wave; k < 28; k += NWAVES) {
        v8f a0 = {};
        mm1_k28(Wl + (size_t)k * 896, oth + (size_t)k * LDC, TN, 0, a0);
        st16(cur + (size_t)k * 256, TN, 0, a0);  // rows 10..15 = in-slice padding
      }
    }
    __syncthreads();

    // ---- S3: C^T @ cur, + bias (+ReLU) -> oth, or bias -> global out ----
    if (!last) {
      for (int nt = wave; nt < 28; nt += NWAVES) {
        v8f a0 = {}, a1 = {};
        mm2_k28(CTm, cur, ldY, nt * 16, a0, a1);
        const int lane = tid & 31;
        const int half = lane >> 4;
        const int col = nt * 16 + (lane & 15);
#pragma unroll
        for (int v = 0; v < 8; ++v) {  // rows 0..15
          int l = v + (half << 3);
          oth[l * LDC + col] = fmaxf(a0[v] + Bl[l * 28 + nt], 0.f);
        }
#pragma unroll
        for (int v = 0; v < 4; ++v) {  // rows 16..27 (always valid)
          int l = 16 + (half << 3) + v;
          oth[l * LDC + col] = fmaxf(a1[v] + Bl[l * 28 + nt], 0.f);
        }
#pragma unroll
        for (int v = 4; v < 8; ++v) {  // rows 20..23 valid / 28..31 -> dump
          int l = 16 + (half << 3) + v;
          int lb = (l < 28) ? l : 27;
          float y = fmaxf(a1[v] + Bl[lb * 28 + nt], 0.f);
          float* q = (l < 28) ? (oth + l * LDC + col) : dump;
          *q = y;
        }
      }
    } else {
      for (int nt = wave; nt < 10; nt += NWAVES) {
        v8f a0 = {}, a1 = {};
        mm2_k28(CTm, cur, ldY, nt * 16, a0, a1);
        const int lane = tid & 31;
        const int half = lane >> 4;
        const int n = lane & 15;
        const bool nok = (n0 + n) < N;
#pragma unroll
        for (int v = 0; v < 8; ++v) {  // rows 0..15
          int l = v + (half << 3);
          if (nok) out[(size_t)(l * 10 + nt) * N + n0 + n] = a0[v] + Bl[l * 10 + nt];
        }
#pragma unroll
        for (int v = 0; v < 8; ++v) {  // rows 16..31 (28..31 dropped)
          int l = 16 + (half << 3) + v;
          if (nok && l < 28)
            out[(size_t)(l * 10 + nt) * N + n0 + n] = a1[v] + Bl[l * 10 + nt];
        }
      }
    }
    __syncthreads();

    float* tmp = cur;
    cur = oth;
    oth = tmp;
  }
}

// ---------------- launch ----------------

extern "C" void kernel_launch(void* const* d_in, const int* in_sizes, int n_in,
                              void* d_out, int out_size, void* d_ws, size_t ws_size,
                              hipStream_t stream) {
  const float* x  = (const float*)d_in[0];
  const float* W1 = (const float*)d_in[1];
  const float* B1 = (const float*)d_in[2];
  const float* W2 = (const float*)d_in[3];
  const float* B2 = (const float*)d_in[4];
  const float* W3 = (const float*)d_in[5];
  const float* B3 = (const float*)d_in[6];
  const float* W4 = (const float*)d_in[7];
  const float* B4 = (const float*)d_in[8];
  const int N = in_sizes[0] / (28 * 28);
  float* ws = (float*)d_ws;  // needs ~409 KB: C(896) + CT(896) + 4x What(25088 each)

  init_dct<<<4, 256, 0, stream>>>(ws);
  const int wtot = 28 * 32 * 28;
  dim3 wgrid((wtot + 255) / 256);
  init_what<<<wgrid, 256, 0, stream>>>(ws, W1, ws + 1792, 28);
  init_what<<<wgrid, 256, 0, stream>>>(ws, W2, ws + 26880, 28);
  init_what<<<wgrid, 256, 0, stream>>>(ws, W3, ws + 51968, 28);
  init_what<<<wgrid, 256, 0, stream>>>(ws, W4, ws + 77056, 10);

  const int nblocks = (N + TN - 1) / TN;
  const size_t shmem = (size_t)(2 * ACT_ELEMS + 16) * sizeof(float);  // ~100.4 KB
  tnn_fused<<<nblocks, NTHREADS, shmem, stream>>>(x, ws, B1, B2, B3, B4,
                                                  (float*)d_out, N);
}